// c_edge_89223650607169
// MI455X (gfx1250) — compile-verified
//
#include <hip/hip_runtime.h>
#include <stdint.h>

// Canny edge detection, fused, for MI455X (gfx1250, wave32).
//
//  Kernel A (canny_front): x -> separable blur -> sobel -> G^2 -> NMS ->
//            packed 1-bit strong/weak masks (ballot_w32, 32 px/word).
//            Tile staging uses CDNA5 async global->LDS (b128 fast path).
//  Kernel B (canny_hyst):  exact 20-iteration hysteresis via radius-20 halo
//            in LDS (bitwise dilation), expand bits -> 0/255 f32.
//
// Traffic: 128MiB read (x) + 8MiB masks + 128MiB write (out) ~ 280MB ~ 13us
// at 23.3 TB/s. No matmul structure -> WMMA not applicable; async-tensor path
// + wave32 ballot are the CDNA5 features that pay here.

#define IMG_W 2048
#define IMG_H 2048
#define NIMG  8
#define ROW_WORDS 64            // 2048 / 32 bits
#define TAN225 0.41421356237f
#define TAN675 2.41421356237f

#define CANNY_USE_ASYNC_LDS 1

// jnp.pad 'reflect' (no edge repeat): -1 -> 1, 2048 -> 2046. Valid for |halo|<W.
__device__ __forceinline__ int reflect2048(int i) {
    i = (i < 0) ? -i : i;
    i = (i >= 2048) ? (4094 - i) : i;
    return i;
}

__device__ __forceinline__ unsigned ballot32(bool p) {
#if __has_builtin(__builtin_amdgcn_ballot_w32)
    return __builtin_amdgcn_ballot_w32(p);
#else
    return (unsigned)__ballot((int)p);
#endif
}

__device__ __forceinline__ void async_load_b32_to_lds(const float* src, float* ldsdst) {
#if CANNY_USE_ASYNC_LDS
    unsigned lds_off = (unsigned)(uintptr_t)ldsdst;   // low 32b of generic ptr = LDS offset
    unsigned long long ga = (unsigned long long)(uintptr_t)src;
    asm volatile("global_load_async_to_lds_b32 %0, %1, off"
                 :: "v"(lds_off), "v"(ga) : "memory");
#else
    *ldsdst = *src;
#endif
}

__device__ __forceinline__ void async_load_b128_to_lds(const float* src, float* ldsdst) {
#if CANNY_USE_ASYNC_LDS
    unsigned lds_off = (unsigned)(uintptr_t)ldsdst;
    unsigned long long ga = (unsigned long long)(uintptr_t)src;
    asm volatile("global_load_async_to_lds_b128 %0, %1, off"
                 :: "v"(lds_off), "v"(ga) : "memory");
#else
    ((float4*)ldsdst)[0] = ((const float4*)src)[0];
#endif
}

__device__ __forceinline__ void async_wait_all() {
#if CANNY_USE_ASYNC_LDS
    asm volatile("s_wait_asynccnt 0" ::: "memory");
#endif
}

// ---------------------------------------------------------------------------
// Kernel A: fused blur + sobel + NMS + threshold -> packed bit masks
// Tile: 32x32 outputs per 256-thread block (8 wave32s). Halo 3.
// x tile layout: 38 rows x 40 cols (col pad so b128 path covers x0-4..x0+35,
// 16B aligned). Logical halo col c in [0,38) lives at index c+1.
// ---------------------------------------------------------------------------
__global__ __launch_bounds__(256) void canny_front_kernel(
    const float* __restrict__ x,
    unsigned* __restrict__ strongp,
    unsigned* __restrict__ weakp)
{
    __shared__ __align__(16) float xs[38 * 40];  // x tile (stride 40)
    __shared__ float tv[36 * 38];                // vertical gauss pass
    __shared__ float xb[36 * 36];                // blurred, halo 2
    __shared__ float g2[34 * 34];                // gx^2+gy^2+1e-12 (0 outside image)
    __shared__ unsigned char dir[32 * 32];       // NMS direction code, center only

    const int t   = threadIdx.x;
    const int bx  = blockIdx.x, by = blockIdx.y, img = blockIdx.z;
    const int x0  = bx * 32, y0 = by * 32;
    const float* xim = x + (size_t)img * IMG_W * IMG_H;

    // ---- Stage 1: async stage x tile into LDS ----
    if (bx != 0 && bx != 63) {
        // Column-interior: rows may reflect (uniform per row), cols contiguous.
        // 38 rows x 10 b128 chunks covering global cols [x0-4, x0+36).
        for (int it = 0; it < 2; ++it) {
            int idx = t + it * 256;
            idx = idx < 379 ? idx : 379;              // keep EXEC full; dup is benign
            int r = idx / 10, j = idx - r * 10;
            int gy = reflect2048(y0 + r - 3);
            const float* src = xim + (size_t)gy * IMG_W + (x0 - 4) + j * 4;
            async_load_b128_to_lds(src, &xs[r * 40 + j * 4]);
        }
    } else {
        // Column-border: per-lane reflect gather, 38x38 cells at col idx c+1.
        for (int it = 0; it < 6; ++it) {
            int idx = t + it * 256;
            idx = idx < 1443 ? idx : 1443;
            int r = idx / 38, c = idx - r * 38;
            int gy = reflect2048(y0 + r - 3);
            int gx = reflect2048(x0 + c - 3);
            async_load_b32_to_lds(xim + (size_t)gy * IMG_W + gx, &xs[r * 40 + c + 1]);
        }
    }
    async_wait_all();
    __syncthreads();

    // float32 Gaussian(sigma=1) weights: [a, b, a]
    const float ga = 0.27406862f, gb = 0.45186276f;

    // ---- Stage 2: vertical blur -> tv (36 rows x 38 cols) ----
    // tv[r][c] = blur_v at global (y0-2+r, x0-3+c); xs col index = c+1.
    for (int idx = t; idx < 36 * 38; idx += 256) {
        int r = idx / 38, c = idx - r * 38;
        const float* p = &xs[r * 40 + c + 1];
        tv[idx] = ga * (p[0] + p[80]) + gb * p[40];
    }
    __syncthreads();

    // ---- Stage 3: horizontal blur -> xb (36 x 36) ----
    for (int idx = t; idx < 36 * 36; idx += 256) {
        int r = idx / 36, c = idx - r * 36;
        const float* p = &tv[r * 38 + c];
        xb[idx] = ga * (p[0] + p[2]) + gb * p[1];
    }
    __syncthreads();

    // ---- Stage 4: G^2 (34 x 34) + direction code for the 32x32 center ----
    for (int idx = t; idx < 34 * 34; idx += 256) {
        int r = idx / 34, c = idx - r * 34;
        int gyc = y0 + r - 1, gxc = x0 + c - 1;
        float v = 0.0f;
        if (gyc >= 0 && gyc < IMG_H && gxc >= 0 && gxc < IMG_W) {
            const float* b = &xb[r * 36 + c];
            float sgx = (b[2] - b[0]) + 2.0f * (b[38] - b[36]) + (b[74] - b[72]);
            float sgy = (b[72] - b[0]) + 2.0f * (b[73] - b[1]) + (b[74] - b[2]);
            v = sgx * sgx + sgy * sgy + 1e-12f;
            if (r >= 1 && r < 33 && c >= 1 && c < 33) {
                float agx = fabsf(sgx), agy = fabsf(sgy);
                unsigned d;
                if (agy < TAN225 * agx)        d = 0;  // d0:   E/W
                else if (agy > TAN675 * agx)   d = 1;  // d90:  N/S
                else d = ((sgx > 0.0f) == (sgy > 0.0f)) ? 2u : 3u;  // d45 / d135
                dir[(r - 1) * 32 + (c - 1)] = (unsigned char)d;
            }
        }
        g2[idx] = v;
    }
    __syncthreads();

    // ---- Stage 5: NMS + thresholds; lane l = pixel x0+l; ballot packs word ----
    const int qdr[4] = { 0, -1, -1, -1 }, qdc[4] = { 1, 0, 1, -1 };
    const int rdr[4] = { 0,  1,  1,  1 }, rdc[4] = {-1, 0,-1,  1 };
    const int wv = t >> 5, l = t & 31;
    for (int rr = wv; rr < 32; rr += 8) {
        int d = (int)dir[rr * 32 + l];
        float c2 = g2[(rr + 1) * 34 + (l + 1)];
        float q2 = g2[(rr + 1 + qdr[d]) * 34 + (l + 1 + qdc[d])];
        float r2 = g2[(rr + 1 + rdr[d]) * 34 + (l + 1 + rdc[d])];
        bool pass = (c2 >= q2) && (c2 >= r2);
        bool sb = pass && (c2 > 400.0f);                       // G > 20
        bool wb = pass && (c2 > 100.0f) && (c2 < 400.0f);      // 10 < G < 20
        unsigned sm = ballot32(sb);
        unsigned wm = ballot32(wb);
        if (l == 0) {
            size_t off = ((size_t)img * IMG_H + (y0 + rr)) * ROW_WORDS + bx;
            strongp[off] = sm;
            weakp[off]   = wm;
        }
    }
}

// ---------------------------------------------------------------------------
// Kernel B: exact 20-iteration hysteresis via radius-20 halo in LDS (bitwise),
// then expand to 0/255 float output. Center: 4 words (128px) x 64 rows;
// region: 6 words x 104 rows. Light-cone: center cells exact after 20 iters.
// ---------------------------------------------------------------------------
#define RW_ 6
#define RH_ 104

__global__ __launch_bounds__(256) void canny_hyst_kernel(
    const unsigned* __restrict__ strongp,
    const unsigned* __restrict__ weakp,
    float* __restrict__ out)
{
    __shared__ unsigned sbuf[RH_ * RW_];
    __shared__ unsigned hbuf[RH_ * RW_];
    __shared__ unsigned wbuf[RH_ * RW_];

    const int t  = threadIdx.x;
    const int bx = blockIdx.x, by = blockIdx.y, img = blockIdx.z;
    const int W0 = bx * 4;          // first center word
    const int y0 = by * 64;         // first center row

    // Load region masks (zero outside image = reference's zero-pad dilation).
    for (int idx = t; idx < RH_ * RW_; idx += 256) {
        int r = idx / RW_, w = idx - r * RW_;
        int gy = y0 - 20 + r;
        int gw = W0 - 1 + w;
        unsigned sv = 0u, wv = 0u;
        if (gy >= 0 && gy < IMG_H && gw >= 0 && gw < ROW_WORDS) {
            size_t off = ((size_t)img * IMG_H + gy) * ROW_WORDS + gw;
            sv = strongp[off];
            wv = weakp[off];
        }
        sbuf[idx] = sv;
        wbuf[idx] = wv;
    }
    __syncthreads();

    for (int iter = 0; iter < 20; ++iter) {
        // Phase 1: horizontal dilate (with cross-word carry bits).
        for (int idx = t; idx < RH_ * RW_; idx += 256) {
            int w = idx % RW_;
            unsigned cv = sbuf[idx];
            unsigned lw = (w > 0)       ? sbuf[idx - 1] : 0u;
            unsigned rw = (w < RW_ - 1) ? sbuf[idx + 1] : 0u;
            hbuf[idx] = cv | (cv << 1) | (cv >> 1) | (lw >> 31) | (rw << 31);
        }
        __syncthreads();
        // Phase 2: vertical OR of 3 rows, gate by weak, accumulate into strong.
        for (int idx = t; idx < RH_ * RW_; idx += 256) {
            int r = idx / RW_;
            unsigned v = hbuf[idx];
            if (r > 0)       v |= hbuf[idx - RW_];
            if (r < RH_ - 1) v |= hbuf[idx + RW_];
            sbuf[idx] |= (wbuf[idx] & v);
        }
        __syncthreads();
    }

    // Expand center bits -> 0/255 floats; lane = bit -> fully coalesced stores.
    const int wv_ = t >> 5, l = t & 31;
    for (int k = wv_; k < 4 * 64; k += 8) {
        int c  = k & 3;       // center word 0..3
        int ry = k >> 2;      // center row 0..63
        unsigned bits = sbuf[(ry + 20) * RW_ + (c + 1)];
        float val = ((bits >> l) & 1u) ? 255.0f : 0.0f;
        size_t off = ((size_t)img * IMG_H + (y0 + ry)) * IMG_W
                   + (size_t)(W0 + c) * 32 + l;
        out[off] = val;
    }
}

// ---------------------------------------------------------------------------
extern "C" void kernel_launch(void* const* d_in, const int* in_sizes, int n_in,
                              void* d_out, int out_size, void* d_ws, size_t ws_size,
                              hipStream_t stream) {
    (void)in_sizes; (void)n_in; (void)out_size; (void)ws_size;
    const float* x = (const float*)d_in[0];
    float* out = (float*)d_out;

    // Workspace: two bit-planes, 8 images * 2048 rows * 64 words * 4B = 4MiB each.
    unsigned* strongp = (unsigned*)d_ws;
    unsigned* weakp   = strongp + (size_t)NIMG * IMG_H * ROW_WORDS;

    dim3 gA(IMG_W / 32, IMG_H / 32, NIMG);
    canny_front_kernel<<<gA, 256, 0, stream>>>(x, strongp, weakp);

    dim3 gB(IMG_W / 128, IMG_H / 64, NIMG);
    canny_hyst_kernel<<<gB, 256, 0, stream>>>(strongp, weakp, out);
}